// KGATConv_43550968382011
// MI455X (gfx1250) — compile-verified
//
#include <hip/hip_runtime.h>
#include <math.h>

// CDNA5 / gfx1250: wave32, WMMA fp32 path (V_WMMA_F32_16X16X4_F32).
typedef float v2f __attribute__((ext_vector_type(2)));
typedef float v4f __attribute__((ext_vector_type(4)));
typedef float v8f __attribute__((ext_vector_type(8)));

#define NEG_INF (-3.402823466e38f)

// ---------------------------------------------------------------- utilities
__global__ __launch_bounds__(256) void zero_f32_kernel(float* p, int n) {
  int i = blockIdx.x * blockDim.x + threadIdx.x;
  if (i < n) p[i] = 0.0f;
}

__global__ __launch_bounds__(32) void zero_i32_kernel(int* p, int n) {
  int i = threadIdx.x;
  if (i < n) p[i] = 0;
}

// ------------------------------------------------------------- B repacking
// Rearrange the 16 relation matrices (and W as matrix #16) from row-major
// [32 in][32 out] into the V_WMMA_F32_16X16X4_F32 B-fragment lane layout so
// each lane reads its per-chunk B values with one global_load_b128:
//   packed[((m*8 + c)*32 + lane)*4 + {0..3}] =
//     { B[k0][col], B[k0+1][col], B[k0][col+16], B[k0+1][col+16] }
//   with col = lane&15, k0 = 4c + 2*(lane>>4).
__global__ __launch_bounds__(256) void pack_b_kernel(
    const float* __restrict__ rel, const float* __restrict__ W,
    float* __restrict__ packed) {
  int idx = blockIdx.x * blockDim.x + threadIdx.x;   // 17 * 8 * 32 total
  if (idx >= 17 * 8 * 32) return;
  int lane = idx & 31;
  int c    = (idx >> 5) & 7;
  int m    = idx >> 8;
  const float* B = (m < 16) ? (rel + (size_t)m * 1024) : W;
  int col  = lane & 15;
  int half = lane >> 4;
  int k0   = 4 * c + 2 * half;
  float* o = packed + (size_t)idx * 4;
  o[0] = B[(size_t)k0 * 32 + col];
  o[1] = B[(size_t)(k0 + 1) * 32 + col];
  o[2] = B[(size_t)k0 * 32 + 16 + col];
  o[3] = B[(size_t)(k0 + 1) * 32 + 16 + col];
}

// ---------------------------------------------------------------- h = x @ W
// One wave per 16-row tile. A = x[16 rows][32], B = packed W fragments.
__global__ __launch_bounds__(32) void gemm_h_kernel(
    const float* __restrict__ x, const float* __restrict__ packedW,
    float* __restrict__ h, int Nn) {
  int lane = threadIdx.x;
  int M = lane & 15;
  int half = lane >> 4;
  int row0 = blockIdx.x * 16;

  int rowA = row0 + M;
  bool rv = rowA < Nn;
  int rowC = rv ? rowA : (Nn - 1);                 // clamped, always valid
  const v2f* xrow2 = (const v2f*)(x + (size_t)rowC * 32);

  v8f d0 = {};
  v8f d1 = {};
#pragma unroll
  for (int c = 0; c < 8; ++c) {
    v2f av = xrow2[2 * c + half];                  // {x[k0], x[k0+1]}
    v2f a;
    a.x = rv ? av.x : 0.0f;                        // v_cndmask, no branch
    a.y = rv ? av.y : 0.0f;
    v4f bv = *(const v4f*)(packedW + ((size_t)c * 32 + lane) * 4);
    v2f b0; b0.x = bv.x; b0.y = bv.y;
    v2f b1; b1.x = bv.z; b1.y = bv.w;
    d0 = __builtin_amdgcn_wmma_f32_16x16x4_f32(false, a, false, b0, (short)0, d0, false, false);
    d1 = __builtin_amdgcn_wmma_f32_16x16x4_f32(false, a, false, b1, (short)0, d1, false, false);
  }
#pragma unroll
  for (int vr = 0; vr < 8; ++vr) {
    int rr = row0 + vr + 8 * half;   // D: M = vgpr + 8*(lane/16), N = lane%16
    if (rr < Nn) {
      h[(size_t)rr * 32 + M]      = d0[vr];
      h[(size_t)rr * 32 + 16 + M] = d1[vr];
    }
  }
}

// ------------------------------------------------- counting sort by relation
__global__ __launch_bounds__(256) void hist_kernel(
    const int* __restrict__ etype, int* __restrict__ binCount, int E_) {
  __shared__ int cnt[16];
  if (threadIdx.x < 16) cnt[threadIdx.x] = 0;
  __syncthreads();
  for (int i = blockIdx.x * blockDim.x + threadIdx.x; i < E_;
       i += gridDim.x * blockDim.x)
    atomicAdd(&cnt[etype[i] & 15], 1);
  __syncthreads();
  if (threadIdx.x < 16 && cnt[threadIdx.x] != 0)
    atomicAdd(&binCount[threadIdx.x], cnt[threadIdx.x]);
}

__global__ __launch_bounds__(32) void scan_kernel(
    const int* __restrict__ binCount, int* __restrict__ binStart,
    int* __restrict__ binCur, int* __restrict__ batchPre, int E_) {
  if (threadIdx.x == 0) {
    int run = 0, runb = 0;
    for (int r = 0; r < 16; ++r) {
      binStart[r] = run;
      binCur[r] = run;
      batchPre[r] = runb;
      run += binCount[r];
      runb += (binCount[r] + 15) >> 4;
    }
    binStart[16] = run;     // == E_
    batchPre[16] = runb;    // total number of 16-edge batches
  }
}

__global__ __launch_bounds__(256) void scatter_kernel(
    const int* __restrict__ etype, int* __restrict__ binCur,
    int* __restrict__ sorted, int E_) {
  __shared__ int cnt[16];
  __shared__ int base[16];
  __shared__ int cur[16];
  int tid = threadIdx.x;
  if (tid < 16) { cnt[tid] = 0; cur[tid] = 0; }
  __syncthreads();
  int e = blockIdx.x * blockDim.x + tid;
  int t = -1;
  if (e < E_) {
    t = etype[e] & 15;
    atomicAdd(&cnt[t], 1);
  }
  __syncthreads();
  if (tid < 16 && cnt[tid] > 0) base[tid] = atomicAdd(&binCur[tid], cnt[tid]);
  __syncthreads();
  if (t >= 0) {
    int r = atomicAdd(&cur[t], 1);
    sorted[base[t] + r] = e;
  }
}

// --------------------------------- per-edge matvec via WMMA + atomic scatter
// One wave per batch of 16 same-relation sorted edges.
// A = gathered h rows [16x32], B = packed rel fragments, D = msgs [16x32].
__global__ __launch_bounds__(32) void msg_kernel(
    const float* __restrict__ h, const int* __restrict__ edge_index,
    const float* __restrict__ packedB, const int* __restrict__ sorted,
    const int* __restrict__ binStart, const int* __restrict__ batchPre,
    float* __restrict__ acc, int E_) {
  int batch = blockIdx.x;
  int total = batchPre[16];
  if (batch >= total) return;           // uniform branch, EXEC stays full

  // locate relation bin for this batch (scalar, 16 bins)
  int r = 0;
#pragma unroll
  for (int i = 1; i < 16; ++i)
    if (batch >= batchPre[i]) r = i;
  int lb = batch - batchPre[r];
  int eStart = binStart[r] + 16 * lb;
  int cnt = binStart[r + 1] - eStart;
  if (cnt > 16) cnt = 16;

  int lane = threadIdx.x;
  int M = lane & 15;
  int half = lane >> 4;
  bool rowValid = (M < cnt);
  int Mc = rowValid ? M : (cnt - 1);    // clamped edge slot, always in-range

  int eid = sorted[eStart + Mc];
  int src = edge_index[eid];            // row 0 of (2,E)
  int dst = edge_index[E_ + eid];       // row 1 of (2,E)

  const v2f* hrow2 = (const v2f*)(h + (size_t)src * 32);
  const float* B = packedB + (size_t)r * 1024;   // packed fragments, matrix r

  v8f d0 = {};
  v8f d1 = {};
#pragma unroll
  for (int c = 0; c < 8; ++c) {
    v2f av = hrow2[2 * c + half];       // {h[k0], h[k0+1]}
    v2f a;
    a.x = rowValid ? av.x : 0.0f;       // zero-pad partial batches, no branch
    a.y = rowValid ? av.y : 0.0f;
    v4f bv = *(const v4f*)(B + ((size_t)c * 32 + lane) * 4);
    v2f b0; b0.x = bv.x; b0.y = bv.y;
    v2f b1; b1.x = bv.z; b1.y = bv.w;
    d0 = __builtin_amdgcn_wmma_f32_16x16x4_f32(false, a, false, b0, (short)0, d0, false, false);
    d1 = __builtin_amdgcn_wmma_f32_16x16x4_f32(false, a, false, b1, (short)0, d1, false, false);
  }

  // scatter-add D rows into destination nodes
#pragma unroll
  for (int vr = 0; vr < 8; ++vr) {
    int row = vr + 8 * half;                // edge slot owning this D row
    int drow = __shfl(dst, row, 32);        // dst of that edge (lane 'row' holds it)
    if (row < cnt) {
      atomicAdd(&acc[(size_t)drow * 32 + M], d0[vr]);
      atomicAdd(&acc[(size_t)drow * 32 + 16 + M], d1[vr]);
    }
  }
}

// ----------------------------------------------------- attention softmax path
__global__ __launch_bounds__(256) void scores_kernel(
    const float* __restrict__ acc, const float* __restrict__ att,
    float* __restrict__ scores, int Nn) {
  int n = blockIdx.x * blockDim.x + threadIdx.x;
  if (n < Nn) {
    const v4f* a = (const v4f*)(acc + (size_t)n * 32);
    float s = 0.0f;
#pragma unroll
    for (int d = 0; d < 8; ++d) {
      v4f v = a[d];
      s += v.x * att[4 * d] + v.y * att[4 * d + 1] +
           v.z * att[4 * d + 2] + v.w * att[4 * d + 3];
    }
    scores[n] = s;
  }
}

__global__ __launch_bounds__(256) void red_max_partial(
    const float* __restrict__ scores, float* __restrict__ partial, int Nn) {
  __shared__ float sm[256];
  int tid = threadIdx.x;
  int i = blockIdx.x * 256 + tid;
  sm[tid] = (i < Nn) ? scores[i] : NEG_INF;
  __syncthreads();
  for (int s = 128; s > 0; s >>= 1) {
    if (tid < s) sm[tid] = fmaxf(sm[tid], sm[tid + s]);
    __syncthreads();
  }
  if (tid == 0) partial[blockIdx.x] = sm[0];
}

__global__ __launch_bounds__(512) void red_max_final(
    const float* __restrict__ partial, float* __restrict__ gmax, int P) {
  __shared__ float sm[512];
  int tid = threadIdx.x;
  float v = NEG_INF;
  for (int i = tid; i < P; i += 512) v = fmaxf(v, partial[i]);
  sm[tid] = v;
  __syncthreads();
  for (int s = 256; s > 0; s >>= 1) {
    if (tid < s) sm[tid] = fmaxf(sm[tid], sm[tid + s]);
    __syncthreads();
  }
  if (tid == 0) gmax[0] = sm[0];
}

__global__ __launch_bounds__(256) void exp_partial(
    const float* __restrict__ scores, const float* __restrict__ gmax,
    float* __restrict__ partial, int Nn) {
  __shared__ float sm[256];
  float m = gmax[0];
  int tid = threadIdx.x;
  int i = blockIdx.x * 256 + tid;
  sm[tid] = (i < Nn) ? __expf(scores[i] - m) : 0.0f;
  __syncthreads();
  for (int s = 128; s > 0; s >>= 1) {
    if (tid < s) sm[tid] += sm[tid + s];
    __syncthreads();
  }
  if (tid == 0) partial[blockIdx.x] = sm[0];
}

__global__ __launch_bounds__(512) void red_sum_final(
    const float* __restrict__ partial, float* __restrict__ gsum, int P) {
  __shared__ float sm[512];
  int tid = threadIdx.x;
  float v = 0.0f;
  for (int i = tid; i < P; i += 512) v += partial[i];
  sm[tid] = v;
  __syncthreads();
  for (int s = 256; s > 0; s >>= 1) {
    if (tid < s) sm[tid] += sm[tid + s];
    __syncthreads();
  }
  if (tid == 0) gsum[0] = sm[0];
}

__global__ __launch_bounds__(256) void final_kernel(
    float* __restrict__ out, const float* __restrict__ scores,
    const float* __restrict__ gmax, const float* __restrict__ gsum, int Nn) {
  int idx = blockIdx.x * blockDim.x + threadIdx.x;
  if (idx < Nn * 32) {
    int n = idx >> 5;
    float w = __expf(scores[n] - gmax[0]) / gsum[0];
    float v = out[idx] * w;
    out[idx] = v > 0.0f ? v : 0.0f;
  }
}

// ------------------------------------------------------------------ launcher
extern "C" void kernel_launch(void* const* d_in, const int* in_sizes, int n_in,
                              void* d_out, int out_size, void* d_ws, size_t ws_size,
                              hipStream_t stream) {
  const float* x   = (const float*)d_in[0];
  const int*   ei  = (const int*)d_in[1];
  const int*   et  = (const int*)d_in[2];
  const float* W   = (const float*)d_in[3];
  const float* rel = (const float*)d_in[4];
  const float* att = (const float*)d_in[5];
  float* out = (float*)d_out;

  const int N_ = in_sizes[0] / 32;
  const int E_ = in_sizes[2];

  // workspace carve-out (~15.6 MB total)
  char* ws = (char*)d_ws;
  size_t off = 0;
  auto carve = [&](size_t bytes) -> char* {
    char* p = ws + off;
    off = (off + bytes + 255) & ~(size_t)255;
    return p;
  };
  float* h        = (float*)carve((size_t)N_ * 32 * sizeof(float));
  float* scores   = (float*)carve((size_t)N_ * sizeof(float));
  int*   sorted   = (int*)carve((size_t)E_ * sizeof(int));
  int*   binCount = (int*)carve(16 * sizeof(int));
  int*   binStart = (int*)carve(17 * sizeof(int));
  int*   binCur   = (int*)carve(16 * sizeof(int));
  int*   batchPre = (int*)carve(17 * sizeof(int));
  float* packed   = (float*)carve((size_t)17 * 1024 * sizeof(float)); // 16 rels + W
  const int NB = (N_ + 255) / 256;
  float* partial  = (float*)carve((size_t)NB * sizeof(float));
  float* gmax     = (float*)carve(sizeof(float));
  float* gsum     = (float*)carve(sizeof(float));
  (void)ws_size; (void)n_in; (void)out_size;

  float* packedW = packed + (size_t)16 * 1024;

  // 1. zero accumulator (d_out), histogram bins; pack B fragments
  zero_f32_kernel<<<(N_ * 32 + 255) / 256, 256, 0, stream>>>(out, N_ * 32);
  zero_i32_kernel<<<1, 32, 0, stream>>>(binCount, 16);
  pack_b_kernel<<<17, 256, 0, stream>>>(rel, W, packed);

  // 2. h = x @ W  (WMMA, one wave per 16 rows)
  gemm_h_kernel<<<(N_ + 15) / 16, 32, 0, stream>>>(x, packedW, h, N_);

  // 3-5. counting sort of edges by relation type
  hist_kernel<<<1024, 256, 0, stream>>>(et, binCount, E_);
  scan_kernel<<<1, 32, 0, stream>>>(binCount, binStart, binCur, batchPre, E_);
  scatter_kernel<<<(E_ + 255) / 256, 256, 0, stream>>>(et, binCur, sorted, E_);

  // 6. per-edge relation matvec (WMMA) + atomic scatter-add
  int maxBatches = (E_ / 16) + 16;
  msg_kernel<<<maxBatches, 32, 0, stream>>>(h, ei, packed, sorted, binStart,
                                            batchPre, out, E_);

  // 7-11. global attention softmax
  scores_kernel<<<(N_ + 255) / 256, 256, 0, stream>>>(out, att, scores, N_);
  red_max_partial<<<NB, 256, 0, stream>>>(scores, partial, N_);
  red_max_final<<<1, 512, 0, stream>>>(partial, gmax, NB);
  exp_partial<<<NB, 256, 0, stream>>>(scores, gmax, partial, N_);
  red_sum_final<<<1, 512, 0, stream>>>(partial, gsum, NB);

  // 12. out = relu(acc * softmax_weight)
  final_kernel<<<(N_ * 32 + 255) / 256, 256, 0, stream>>>(out, scores, gmax,
                                                          gsum, N_);
}